// Criterion_32830730011569
// MI455X (gfx1250) — compile-verified
//
#include <hip/hip_runtime.h>
#include <hip/hip_bf16.h>

// ---------------- problem constants (from reference) ----------------
#define BB 4
#define HH 192
#define WW 192
#define QQ 160
#define EE 96
#define MM 96
#define KK 4
#define WIN 7
#define NO_ELECTRON_WEIGHT 0.1f
#define LOG_2PI 1.8378770664093455f

#define NPIX 24                        // pixels staged per dice chunk (double-buffered)
#define DICE_CHUNKS ((HH * WW) / NPIX) // 1536

typedef __attribute__((ext_vector_type(2))) float v2f;
typedef __attribute__((ext_vector_type(8))) float v8f;
typedef __attribute__((ext_vector_type(4))) unsigned int u32x4;
typedef __attribute__((ext_vector_type(4))) int i32x4;
typedef __attribute__((ext_vector_type(8))) int i32x8;

#if __has_builtin(__builtin_amdgcn_tensor_load_to_lds) && __has_builtin(__builtin_amdgcn_s_wait_tensorcnt)
#define HAVE_TDM 1
#endif
#if __has_builtin(__builtin_amdgcn_wmma_f32_16x16x4_f32)
#define HAVE_WMMA4 1
#endif

// ---------------- helpers ----------------
__device__ __forceinline__ float softplusf(float x) {
  // logaddexp(0, x) = max(x,0) + log1p(exp(-|x|))
  return fmaxf(x, 0.f) + log1pf(expf(-fabsf(x)));
}

__device__ __forceinline__ float warpReduceSum(float v) {
  for (int o = 16; o > 0; o >>= 1) v += __shfl_xor(v, o, 32);
  return v;
}

__device__ __forceinline__ float blockReduceSum(float v, float* scr) {
  const int lane = threadIdx.x & 31;
  const int wid = threadIdx.x >> 5;
  v = warpReduceSum(v);
  if (lane == 0) scr[wid] = v;
  __syncthreads();
  const int nw = (blockDim.x + 31) >> 5;
  v = (threadIdx.x < (unsigned)nw) ? scr[threadIdx.x] : 0.f;
  if (wid == 0) v = warpReduceSum(v);
  return v; // valid on thread 0
}

#ifdef HAVE_TDM
// 1-D contiguous DMA: n float32 elements from gaddr -> LDS byte offset lds_off.
// D# per CDNA5 ISA ch.8 (group0: count/lds/global/type, group1: data_size=4B,
// tensor_dim0=tile_dim0=n, tensor_dim1=tile_dim1=1, dim0_stride=n).
__device__ __forceinline__ void tdm_load_1d(unsigned int lds_off,
                                            unsigned long long gaddr,
                                            unsigned int n) {
  u32x4 g0 = {
      1u,                                   // count=1 valid descriptor
      lds_off,                              // lds_addr (bytes)
      (unsigned int)gaddr,                  // global_addr[31:0]
      (((unsigned int)(gaddr >> 32)) & 0x1FFFFFFu) | (2u << 30) // addr[56:32] | type=2
  };
  i32x8 g1 = {
      (int)(2u << 16),                      // data_size = 4B
      (int)((n & 0xFFFFu) << 16),           // tensor_dim0[15:0] @ bits 63:48
      (int)((n >> 16) | (1u << 16)),        // tensor_dim0[31:16] | tensor_dim1=1
      (int)((n & 0xFFFFu) << 16),           // tile_dim0 @ bits 127:112
      1,                                    // tile_dim1 = 1
      (int)n,                               // tensor_dim0_stride[31:0]
      0, 0
  };
  i32x4 g2 = {0, 0, 0, 0};
  i32x4 g3 = {0, 0, 0, 0};
#if defined(__clang_major__) && (__clang_major__ >= 23)
  i32x8 g4 = {0, 0, 0, 0, 0, 0, 0, 0};
  __builtin_amdgcn_tensor_load_to_lds(g0, g1, g2, g3, g4, 0);
#else
  __builtin_amdgcn_tensor_load_to_lds(g0, g1, g2, g3, 0);
#endif
}
#endif

// ---------------- workspace layout (floats) ----------------
// ws[0..3]  : dice numerator per batch
// ws[4..7]  : dice denominator per batch
// ws[8]     : class loss
// ws[9]     : mask-window BCE loss
// ws[10]    : NLL loss
// ws[11]    : occupancy loss

__global__ void zero_kernel(float* ws, float* out) {
  const int t = threadIdx.x;
  if (t < 16) ws[t] = 0.f;
  if (t == 0) out[0] = 0.f;
}

// ---------------- class loss ----------------
__global__ void class_loss_kernel(const float* __restrict__ logit,
                                  const int* __restrict__ mq, float* ws) {
  __shared__ float labels[BB * QQ];
  __shared__ float scr[32];
  const int t = threadIdx.x;
  for (int i = t; i < BB * QQ; i += blockDim.x) labels[i] = 0.f;
  __syncthreads();
  if (t < BB * MM) {
    const int b = t / MM, m = t % MM;
    labels[b * QQ + mq[b * MM + m]] = 1.f; // matched_q unique per batch
  }
  __syncthreads();
  float v = 0.f;
  for (int i = t; i < BB * QQ; i += blockDim.x) {
    const float x = logit[i], z = labels[i];
    const float w = (z > 0.f) ? 1.f : NO_ELECTRON_WEIGHT;
    v += w * (softplusf(x) - x * z);
  }
  const float s = blockReduceSum(v, scr);
  if (t == 0) atomicAdd(&ws[8], s * (1.f / (float)(BB * QQ)));
}

// ---------------- 7x7-window mask BCE ----------------
__global__ void bce_window_kernel(const float* __restrict__ tseg,
                                  const float* __restrict__ bml,
                                  const float* __restrict__ inc,
                                  const int* __restrict__ mq,
                                  const int* __restrict__ me, float* ws) {
  __shared__ float scr[32];
  const int bm = blockIdx.x;           // [0, B*M)
  const int b = bm / MM, m = bm % MM;
  const int e = me[b * MM + m];
  const int q = mq[b * MM + m];
  const float py = inc[((size_t)b * EE + e) * 2 + 0];
  const float px = inc[((size_t)b * EE + e) * 2 + 1];
  const int pr = (int)floorf(py);
  const int pc = (int)floorf(px);
  const int t = threadIdx.x;
  float v = 0.f;
  if (t < WIN * WIN) {
    const int r = pr + (t / WIN) - (WIN / 2);
    const int c = pc + (t % WIN) - (WIN / 2);
    const float tv = tseg[(((size_t)b * HH + r) * WW + c) * EE + e];
    const float lg = bml[(((size_t)b * HH + r) * WW + c) * QQ + q];
    v = softplusf(lg) - lg * tv;
  }
  const float s = blockReduceSum(v, scr);
  if (t == 0) atomicAdd(&ws[9], s * (1.f / (float)(BB * MM * WIN * WIN)));
}

// ---------------- distance NLL ----------------
__global__ void nll_kernel(const float* __restrict__ inc,
                           const float* __restrict__ pos,
                           const float* __restrict__ chol,
                           const int* __restrict__ mq,
                           const int* __restrict__ me, float* ws) {
  __shared__ float scr[32];
  const int i = threadIdx.x;
  float v = 0.f;
  if (i < BB * MM) {
    const int b = i / MM;
    const int e = me[i];
    const int q = mq[i];
    const float p0 = inc[((size_t)b * EE + e) * 2 + 0];
    const float p1 = inc[((size_t)b * EE + e) * 2 + 1];
    const float c0 = pos[((size_t)b * QQ + q) * 2 + 0];
    const float c1 = pos[((size_t)b * QQ + q) * 2 + 1];
    const float* ch = chol + ((size_t)b * QQ + q) * 4; // [2][2] row-major
    const float l00 = ch[0], l10 = ch[2], l11 = ch[3];
    const float d0 = p0 - c0, d1 = p1 - c1;
    const float z0 = d0 / l00;
    const float z1 = (d1 - l10 * z0) / l11;
    float n = 0.5f * (z0 * z0 + z1 * z1) + LOG_2PI + logf(l00) + logf(l11);
    if (isinf(n)) n = 1e7f;
    v = n;
  }
  const float s = blockReduceSum(v, scr);
  if (i == 0) atomicAdd(&ws[10], s * (1.f / (float)(BB * MM)));
}

// ---------------- occupancy cross-entropy (K=4, fused logsumexp) -------------
__global__ void occ_kernel(const float* __restrict__ occl,
                           const int* __restrict__ occt, float* ws) {
  __shared__ float scr[32];
  const size_t pix = (size_t)blockIdx.x * blockDim.x + threadIdx.x; // < B*H*W
  const float4 l4 = ((const float4*)occl)[pix];
  const float mx = fmaxf(fmaxf(l4.x, l4.y), fmaxf(l4.z, l4.w));
  const float se = __expf(l4.x - mx) + __expf(l4.y - mx) +
                   __expf(l4.z - mx) + __expf(l4.w - mx);
  const float lse = mx + logf(se);
  const int tr = occt[pix];
  const float xt = (tr == 0) ? l4.x : (tr == 1) ? l4.y : (tr == 2) ? l4.z : l4.w;
  const float s = blockReduceSum(lse - xt, scr);
  if (threadIdx.x == 0)
    atomicAdd(&ws[11], s * (1.f / (float)(BB * HH * WW)));
}

// ---------------- dice: double-buffered TDM streaming + LDS gather -----------
// TDM DMA for chunk i+1 is in flight while chunk i is consumed; wave 0 waits
// only for TENSORcnt<=2 (in-order completion => older pair has landed).
__global__ void dice_kernel(const float* __restrict__ tseg,
                            const float* __restrict__ pol,
                            const int* __restrict__ mqg,
                            const int* __restrict__ meg, float* ws) {
  __shared__ float s_por[2][NPIX * QQ]; // 2*24*160*4 = 30720 B
  __shared__ float s_tru[2][NPIX * EE]; // 2*24* 96*4 = 18432 B
  __shared__ int s_mq[MM];
  __shared__ int s_me[MM];

  const int t = threadIdx.x;
  const int b = blockIdx.y;
  if (t < MM) {
    s_mq[t] = mqg[b * MM + t];
    s_me[t] = meg[b * MM + t];
  }
  __syncthreads();

  const int lane = t & 31;
  const int wid = t >> 5; // 8 waves of 32 in a 256 block
  // hoist this lane's channel indices (m = lane, lane+32, lane+64)
  const int q0 = s_mq[lane], q1 = s_mq[lane + 32], q2 = s_mq[lane + 64];
  const int e0i = s_me[lane], e1i = s_me[lane + 32], e2i = s_me[lane + 64];

  float numacc = 0.f, denacc = 0.f;

  int c = blockIdx.x;
  bool have = (c < DICE_CHUNKS);
  int buf = 0;

#ifdef HAVE_TDM
  if (have && t < 32) { // prologue: stage first chunk into buffer 0
    const size_t pb = (size_t)b * HH * WW + (size_t)c * NPIX;
    tdm_load_1d((unsigned int)(uintptr_t)&s_por[0][0],
                (unsigned long long)(uintptr_t)(pol + pb * QQ), NPIX * QQ);
    tdm_load_1d((unsigned int)(uintptr_t)&s_tru[0][0],
                (unsigned long long)(uintptr_t)(tseg + pb * EE), NPIX * EE);
  }
#endif

  while (have) {
    const int cn = c + gridDim.x;
    const bool haveN = (cn < DICE_CHUNKS);
#ifdef HAVE_TDM
    if (t < 32) {
      if (haveN) { // stage next chunk into the other buffer while we compute
        const size_t pb = (size_t)b * HH * WW + (size_t)cn * NPIX;
        tdm_load_1d((unsigned int)(uintptr_t)&s_por[buf ^ 1][0],
                    (unsigned long long)(uintptr_t)(pol + pb * QQ), NPIX * QQ);
        tdm_load_1d((unsigned int)(uintptr_t)&s_tru[buf ^ 1][0],
                    (unsigned long long)(uintptr_t)(tseg + pb * EE), NPIX * EE);
        __builtin_amdgcn_s_wait_tensorcnt(2); // current buffer's pair complete
      } else {
        __builtin_amdgcn_s_wait_tensorcnt(0); // drain
      }
    }
#else
    {
      const size_t pb = (size_t)b * HH * WW + (size_t)c * NPIX;
      const float4* gp = (const float4*)(pol + pb * QQ);
      float4* sp4 = (float4*)&s_por[buf][0];
      for (int i = t; i < (NPIX * QQ) / 4; i += blockDim.x) sp4[i] = gp[i];
      const float4* gt = (const float4*)(tseg + pb * EE);
      float4* st4 = (float4*)&s_tru[buf][0];
      for (int i = t; i < (NPIX * EE) / 4; i += blockDim.x) st4[i] = gt[i];
    }
#endif
    __syncthreads(); // current buffer visible to all waves

    for (int k = 0; k < NPIX / 8; ++k) { // 3 pixels per wave
      const int pp = wid * (NPIX / 8) + k;
      const float* prow = &s_por[buf][pp * QQ];
      const float* trow = &s_tru[buf][pp * EE];
      const float p0 = prow[q0], p1 = prow[q1], p2 = prow[q2];
      const float t0 = trow[e0i], t1 = trow[e1i], t2 = trow[e2i];
      // softmax over the 96 matched channels (spread across 32 lanes x 3)
      float mx = fmaxf(p0, fmaxf(p1, p2));
      for (int o = 16; o > 0; o >>= 1) mx = fmaxf(mx, __shfl_xor(mx, o, 32));
      const float x0 = __expf(p0 - mx);
      const float x1 = __expf(p1 - mx);
      const float x2 = __expf(p2 - mx);
      float se = x0 + x1 + x2;
      for (int o = 16; o > 0; o >>= 1) se += __shfl_xor(se, o, 32);
      const float inv = 1.0f / se;
      numacc += 2.f * inv * (t0 * x0 + t1 * x1 + t2 * x2);
      denacc += (t0 + t1 + t2) + inv * (x0 + x1 + x2);
    }

    __syncthreads(); // all waves done with `buf` before it is re-staged
    c = cn;
    have = haveN;
    buf ^= 1;
  }

  numacc = warpReduceSum(numacc);
  denacc = warpReduceSum(denacc);
  if (lane == 0) {
    atomicAdd(&ws[b], numacc);
    atomicAdd(&ws[4 + b], denacc);
  }
}

// ---------------- finalize: combine terms with V_WMMA_F32_16X16X4_F32 --------
// A-matrix row0 = per-batch dice terms (x0.25), row1 = the 4 scalar terms.
// B = all-ones (layout-invariant), C = 0 -> D[0][0]+D[1][0] = total loss.
__global__ void finalize_kernel(const float* __restrict__ ws, float* out) {
  const int lane = threadIdx.x; // exactly 32 threads, EXEC all-1s
  const float w0 = ws[0], w1 = ws[1], w2 = ws[2], w3 = ws[3];
  const float w4 = ws[4], w5 = ws[5], w6 = ws[6], w7 = ws[7];
  const float cls = ws[8], bce = ws[9], nll = ws[10], occ = ws[11];
  const float d0 = 0.25f * (1.f - (w0 + 1.f) / (w4 + 1.f));
  const float d1 = 0.25f * (1.f - (w1 + 1.f) / (w5 + 1.f));
  const float d2 = 0.25f * (1.f - (w2 + 1.f) / (w6 + 1.f));
  const float d3 = 0.25f * (1.f - (w3 + 1.f) / (w7 + 1.f));
#ifdef HAVE_WMMA4
  // branch-free A build so EXEC stays full at the WMMA
  const float f0 = (float)(lane == 0);
  const float f16_ = (float)(lane == 16);
  const float f1 = (float)(lane == 1);
  const float f17 = (float)(lane == 17);
  v2f a;
  a.x = d0 * f0 + d2 * f16_ + cls * f1 + nll * f17; // K even slots
  a.y = d1 * f0 + d3 * f16_ + bce * f1 + occ * f17; // K odd slots
  v2f bvec;
  bvec.x = 1.f;
  bvec.y = 1.f;
  v8f cacc = {};
  v8f dres = __builtin_amdgcn_wmma_f32_16x16x4_f32(
      false, a, false, bvec, (short)0, cacc, false, false);
  float total = dres[0] + dres[1]; // lane0: D[0][0] + D[1][0]
  asm volatile("" : "+v"(total));  // keep WMMA executed wave-wide
  if (lane == 0) out[0] = total;
#else
  if (lane == 0)
    out[0] = d0 + d1 + d2 + d3 + cls + bce + nll + occ;
#endif
}

// ---------------- entry point ----------------
extern "C" void kernel_launch(void* const* d_in, const int* in_sizes, int n_in,
                              void* d_out, int out_size, void* d_ws,
                              size_t ws_size, hipStream_t stream) {
  (void)in_sizes; (void)n_in; (void)out_size; (void)ws_size;
  const float* is_el = (const float*)d_in[0];
  const float* tseg  = (const float*)d_in[1];
  const float* bml   = (const float*)d_in[2];
  const float* pol   = (const float*)d_in[3];
  const float* inc   = (const float*)d_in[4];
  const float* pos   = (const float*)d_in[5];
  const float* chol  = (const float*)d_in[6];
  const float* occl  = (const float*)d_in[7];
  const int*   occt  = (const int*)d_in[8];
  const int*   mq    = (const int*)d_in[9];
  const int*   me    = (const int*)d_in[10];
  float* out = (float*)d_out;
  float* ws  = (float*)d_ws;

  zero_kernel<<<1, 64, 0, stream>>>(ws, out);
  class_loss_kernel<<<1, 1024, 0, stream>>>(is_el, mq, ws);
  bce_window_kernel<<<BB * MM, 64, 0, stream>>>(tseg, bml, inc, mq, me, ws);
  nll_kernel<<<1, 512, 0, stream>>>(inc, pos, chol, mq, me, ws);
  occ_kernel<<<(BB * HH * WW) / 256, 256, 0, stream>>>(occl, occt, ws);
  dice_kernel<<<dim3(192, BB), 256, 0, stream>>>(tseg, pol, mq, me, ws);
  finalize_kernel<<<1, 32, 0, stream>>>(ws, out);
}